// ContrastiveLoss_2405181686310
// MI455X (gfx1250) — compile-verified
//
#include <hip/hip_runtime.h>

typedef __attribute__((ext_vector_type(2))) float v2f;
typedef __attribute__((ext_vector_type(4))) float v4f;
typedef __attribute__((ext_vector_type(8))) float v8f;
typedef __attribute__((ext_vector_type(4))) int   v4i;

#define AS1 __attribute__((address_space(1)))
#define AS3 __attribute__((address_space(3)))

typedef AS1 v4i* gbl4_t;   // global int4*  (async builtin arg0)
typedef AS3 v4i* lds4_t;   // LDS int4*     (async builtin arg1)

#define N_EMB   8192
#define D_EMB   256
#define TILE    16
#define KSTEPS  (D_EMB / 4)            // 64 wmma f32 16x16x4 per column tile
#define WAVES   8
#define ROWS_PER_WG (WAVES * TILE)     // 128
#define NTILES  (N_EMB / TILE)         // 512
#define CSPLIT  4                      // column split -> 256 WGs
#define TILES_PER_WG (NTILES / CSPLIT) // 128
#define LDS_STRIDE 260                 // dwords/row: 260%64==4 -> conflict-free b64 gather
#define LDS_TILE  (TILE * LDS_STRIDE)  // 4160 dwords per buffer (16.6 KB), x2 buffers
#define INV_T   2.0f
#define SHIFT   2.0f                   // == true row max for normalized embeddings

#if __has_builtin(__builtin_amdgcn_global_load_async_to_lds_b128)
#define USE_ASYNC 1
#else
#define USE_ASYNC 0
#endif

__global__ void cl_zero_ws(float* ws) {
    int i = blockIdx.x * blockDim.x + threadIdx.x;
    if (i < 3 * N_EMB) ws[i] = 0.0f;
}

__global__ __launch_bounds__(256) void cl_main(const float* __restrict__ emb,
                                               const int* __restrict__ labels,
                                               float* __restrict__ sArr,
                                               float* __restrict__ pArr,
                                               float* __restrict__ nArr) {
    __shared__ float ldsB[2 * LDS_TILE];

    const int tid  = threadIdx.x;
    const int lane = tid & 31;
    const int wv   = tid >> 5;
    const int hi   = lane >> 4;
    const int ln16 = lane & 15;

    const int rowBlk = blockIdx.x >> 2;          // / CSPLIT
    const int split  = blockIdx.x & (CSPLIT - 1);
    const int jt0    = split * TILES_PER_WG;
    const int jtEnd  = jt0 + TILES_PER_WG;

    // ---- this wave's 16-row A tile resident in registers ------------------
    const int i0   = rowBlk * ROWS_PER_WG + wv * TILE;
    const int arow = i0 + ln16;
    v2f afrag[KSTEPS];                  // A 16x4 f32 layout: lo lanes K+{0,1}, hi lanes K+{2,3}
#pragma unroll
    for (int kk = 0; kk < KSTEPS; ++kk)
        afrag[kk] = *(const v2f*)(emb + (size_t)arow * D_EMB + kk * 4 + hi * 2);

    int rl[8];
#pragma unroll
    for (int r = 0; r < 8; ++r) rl[r] = labels[i0 + r + hi * 8];

    float s_part[8], p_part[8], n_part[8];
#pragma unroll
    for (int r = 0; r < 8; ++r) { s_part[r] = 0.0f; p_part[r] = 0.0f; n_part[r] = 0.0f; }

    // ---- B tile movers (double buffered) ----------------------------------
#if USE_ASYNC
    // global -> LDS directly, tracked by ASYNCcnt
    auto issue = [&](int jt, int buf) {
#pragma unroll
        for (int it = 0; it < 4; ++it) {
            const int q   = tid + it * 256;      // 1024 x b128 chunks per tile
            const int row = q >> 6;
            const int c4  = q & 63;
            const float* g = emb + (size_t)(jt * TILE + row) * D_EMB + c4 * 4;
            float*       l = &ldsB[buf * LDS_TILE + row * LDS_STRIDE + c4 * 4];
            __builtin_amdgcn_global_load_async_to_lds_b128(
                (gbl4_t)g, (lds4_t)l, 0, 0);
        }
    };
    auto finish = [&]() {
#if __has_builtin(__builtin_amdgcn_s_wait_asynccnt)
        __builtin_amdgcn_s_wait_asynccnt(0);
#else
        asm volatile("s_wait_asynccnt 0" ::: "memory");
#endif
    };
#else
    // register-staged fallback, identical LDS layout
    v4f stage[4];
    auto issue = [&](int jt, int) {
#pragma unroll
        for (int it = 0; it < 4; ++it) {
            const int q   = tid + it * 256;
            const int row = q >> 6;
            const int c4  = q & 63;
            stage[it] = *(const v4f*)(emb + (size_t)(jt * TILE + row) * D_EMB + c4 * 4);
        }
    };
    auto flush = [&](int buf) {
#pragma unroll
        for (int it = 0; it < 4; ++it) {
            const int q   = tid + it * 256;
            const int row = q >> 6;
            const int c4  = q & 63;
            *(v4f*)&ldsB[buf * LDS_TILE + row * LDS_STRIDE + c4 * 4] = stage[it];
        }
    };
    auto finish = [&]() {};
#endif

    // ---- preload first tile ----------------------------------------------
    issue(jt0, 0);
#if !USE_ASYNC
    flush(0);
#endif
    finish();
    __syncthreads();

    // ---- main loop: overlap next-tile fill with current-tile WMMA --------
    int cur = 0;
    for (int jt = jt0; jt < jtEnd; ++jt) {
        const int j0      = jt * TILE;
        const bool more   = (jt + 1 < jtEnd);
        if (more) issue(jt + 1, cur ^ 1);

        const int cl  = labels[j0 + ln16];
        const int col = j0 + ln16;
        const float* bbase = &ldsB[cur * LDS_TILE + ln16 * LDS_STRIDE + hi * 2];

        v8f acc = {0.0f, 0.0f, 0.0f, 0.0f, 0.0f, 0.0f, 0.0f, 0.0f};
#pragma unroll
        for (int kk = 0; kk < KSTEPS; ++kk) {
            v2f b = *(const v2f*)(bbase + kk * 4);     // conflict-free gather
            acc = __builtin_amdgcn_wmma_f32_16x16x4_f32(
                false, afrag[kk], false, b, (short)0, acc, false, false);
        }

#pragma unroll
        for (int r = 0; r < 8; ++r) {
            const int rowr  = i0 + r + hi * 8;
            const float sim = acc[r] * INV_T;
            const bool self = (rowr == col);
            s_part[r] += self ? 0.0f : __expf(sim - SHIFT);
            if ((cl == rl[r]) && !self) {
                p_part[r] += sim;
                n_part[r] += 1.0f;
            }
        }

#if !USE_ASYNC
        if (more) flush(cur ^ 1);
#endif
        finish();          // my async fills into buf[cur^1] done
        __syncthreads();   // everyone done: next buffer ready, this one free
        cur ^= 1;
    }

    // ---- half-wave reductions -> per-row atomic partials ------------------
#pragma unroll
    for (int r = 0; r < 8; ++r) {
        float s  = s_part[r];
        float p  = p_part[r];
        float np = n_part[r];
#pragma unroll
        for (int m = 1; m <= 8; m <<= 1) {
            s  += __shfl_xor(s,  m, 32);
            p  += __shfl_xor(p,  m, 32);
            np += __shfl_xor(np, m, 32);
        }
        if (ln16 == 0) {                       // lanes 0 and 16: one per row
            const int row = i0 + r + hi * 8;
            atomicAdd(&sArr[row], s);
            atomicAdd(&pArr[row], p);
            atomicAdd(&nArr[row], np);
        }
    }
}

__global__ __launch_bounds__(256) void cl_finalize(const float* __restrict__ sArr,
                                                   const float* __restrict__ pArr,
                                                   const float* __restrict__ nArr,
                                                   float* __restrict__ out) {
    __shared__ float redC[256], redN[256];
    const int tid = threadIdx.x;
    float c = 0.0f, n = 0.0f;
    for (int i = tid; i < N_EMB; i += 256) {
        const float np = nArr[i];
        c += np * (SHIFT + __logf(sArr[i])) - pArr[i];
        n += np;
    }
    redC[tid] = c; redN[tid] = n;
    __syncthreads();
    for (int off = 128; off > 0; off >>= 1) {
        if (tid < off) { redC[tid] += redC[tid + off]; redN[tid] += redN[tid + off]; }
        __syncthreads();
    }
    if (tid == 0) out[0] = redC[0] / redN[0];
}

extern "C" void kernel_launch(void* const* d_in, const int* in_sizes, int n_in,
                              void* d_out, int out_size, void* d_ws, size_t ws_size,
                              hipStream_t stream) {
    const float* emb    = (const float*)d_in[0];
    const int*   labels = (const int*)d_in[1];
    float*       out    = (float*)d_out;
    float*       ws     = (float*)d_ws;   // [0,8192): s  [8192,16384): possum  [16384,24576): npos

    float* sArr = ws;
    float* pArr = ws + N_EMB;
    float* nArr = ws + 2 * N_EMB;

    cl_zero_ws<<<(3 * N_EMB + 255) / 256, 256, 0, stream>>>(ws);
    cl_main<<<(N_EMB / ROWS_PER_WG) * CSPLIT, WAVES * 32, 0, stream>>>(emb, labels, sArr, pArr, nArr);
    cl_finalize<<<1, 256, 0, stream>>>(sArr, pArr, nArr, out);
}